// MultiPCENlayer_11793980195072
// MI455X (gfx1250) — compile-verified
//
#include <hip/hip_runtime.h>
#include <math.h>

// MultiPCEN filtfilt-EMA + PCEN for MI455X (gfx1250, wave32).
// Scan blocked into 16-step chunks executed as 16x16x16 f32 matmuls via
// 4 chained V_WMMA_F32_16X16X4_F32; forward pass stores only chunk carries,
// backward pass recomputes y and fuses the PCEN elementwise tail.

typedef __attribute__((ext_vector_type(2))) float v2f;
typedef __attribute__((ext_vector_type(8))) float v8f;

#define T_LEN   2000
#define NCHUNK  125          // 125 * 16 == 2000
#define WPB     4            // waves per block
#define NTILES  1024         // 8 (K) * 16 (B) * 8 (F/16)

__device__ __forceinline__ float sigm(float v) { return 1.0f / (1.0f + __expf(-v)); }

__device__ __forceinline__ v8f wmma16x16x16(const v2f* A, v2f b0, v2f b1, v2f b2, v2f b3, v8f c) {
    // D = A(16x16) * B(16x16) + C via 4 chained K=4 f32 WMMAs.
    c = __builtin_amdgcn_wmma_f32_16x16x4_f32(false, A[0], false, b0, (short)0, c, false, false);
    c = __builtin_amdgcn_wmma_f32_16x16x4_f32(false, A[1], false, b1, (short)0, c, false, false);
    c = __builtin_amdgcn_wmma_f32_16x16x4_f32(false, A[2], false, b2, (short)0, c, false, false);
    c = __builtin_amdgcn_wmma_f32_16x16x4_f32(false, A[3], false, b3, (short)0, c, false, false);
    return c;
}

__global__ __launch_bounds__(128) void pcen_wmma_kernel(
    const float* __restrict__ x,
    const float* __restrict__ i_sig_alpha,
    const float* __restrict__ log_delta,
    const float* __restrict__ i_sig_r,
    const float* __restrict__ i_sig_s,
    const float* __restrict__ i_sig_eps,
    float* __restrict__ out)
{
    __shared__ float carries[WPB][NCHUNK + 1][16];  // forward chunk-entry carries
    __shared__ float ztile[WPB][16 * 20];           // D-layout -> B-layout staging (stride 20: conflict-free)

    const int tid  = threadIdx.x;
    const int wave = tid >> 5;
    const int lane = tid & 31;
    const int n    = lane & 15;   // N (row) for B/C/D; M for A
    const int hi   = lane >> 4;

    const int tile = blockIdx.x * WPB + wave;       // 0..1023
    if (tile >= NTILES) return;
    const int k    = tile >> 7;                     // PCEN channel 0..7
    const int rem  = tile & 127;
    const int b    = rem >> 3;                      // batch 0..15
    const int ft   = rem & 7;                       // F-tile 0..7

    const long xrow = (long)(b * 128 + ft * 16 + n) * T_LEN;
    const long orow = (long)((b * 8 + k) * 128 + ft * 16 + n) * T_LEN;

    // ---- per-wave scalar parameters -------------------------------------
    const float alpha = sigm(i_sig_alpha[k]);
    const float delta = __expf(log_delta[k]);
    const float r     = sigm(i_sig_r[k]);
    const float s     = sigm(i_sig_s[k]);
    const float eps   = sigm(i_sig_eps[0]);
    const float a     = 1.0f - s;
    const float lge   = __logf(eps);
    const float inve  = 1.0f / eps;
    const float dr    = __powf(delta, r);
    const float l2a   = log2f(a);                   // a^e = exp2(e*log2 a)

    // ---- chunk recurrence matrices (A-layout: lane=M, {half,vgpr}=K) ----
    v2f Afwd[4], A0m[4], Arev[4], ArevL[4];
    #pragma unroll
    for (int q = 0; q < 4; ++q) {
        #pragma unroll
        for (int v = 0; v < 2; ++v) {
            const int kk = 4 * q + 2 * hi + v;      // column index 0..15
            // interior forward: y_i = sum_{j<=i} s*a^(i-j) x_j  (+ a^(i+1)*carry via C)
            float fw = (n >= kk) ? s * exp2f(l2a * (float)(n - kk)) : 0.0f;
            // interior backward: z_i = sum_{j>=i} s*a^(j-i) y_j (+ a^(16-i)*bcarry via C)
            float rv = (kk >= n) ? s * exp2f(l2a * (float)(kk - n)) : 0.0f;
            // first chunk: y0=x0, y1=x1, carry=x1, scan from t=2
            float f0;
            if (n == 0)      f0 = (kk == 0) ? 1.0f : 0.0f;
            else if (n == 1) f0 = (kk == 1) ? 1.0f : 0.0f;
            else if (kk == 1)            f0 = exp2f(l2a * (float)(n - 1));
            else if (kk >= 2 && kk <= n) f0 = s * exp2f(l2a * (float)(n - kk));
            else                         f0 = 0.0f;
            // last chunk: z15=y15, z14=y14, carry=y14, scan from local t=13 down
            float rl;
            if (n == 15)      rl = (kk == 15) ? 1.0f : 0.0f;
            else if (n == 14) rl = (kk == 14) ? 1.0f : 0.0f;
            else if (kk == 14)             rl = exp2f(l2a * (float)(14 - n));
            else if (kk >= n && kk <= 13)  rl = s * exp2f(l2a * (float)(kk - n));
            else                           rl = 0.0f;
            if (v == 0) { Afwd[q].x = fw; Arev[q].x = rv; A0m[q].x = f0; ArevL[q].x = rl; }
            else        { Afwd[q].y = fw; Arev[q].y = rv; A0m[q].y = f0; ArevL[q].y = rl; }
        }
    }
    // carry coefficients in C/D layout (vgpr j, half -> M = j + 8*hi)
    float cf[8], cr[8];
    #pragma unroll
    for (int j = 0; j < 8; ++j) {
        const int m = j + 8 * hi;
        cf[j] = exp2f(l2a * (float)(m + 1));        // a^(i+1)
        cr[j] = exp2f(l2a * (float)(16 - m));       // a^(16-i)
    }

    // ===================== forward pass: store carries ====================
    float carry_bc = 0.0f;
    for (int c = 0; c < NCHUNK; ++c) {
        const int t0 = c * 16;
        const float* xp = x + xrow + t0 + 2 * hi;
        v2f B0 = *(const v2f*)(xp + 0);
        v2f B1 = *(const v2f*)(xp + 4);
        v2f B2 = *(const v2f*)(xp + 8);
        v2f B3 = *(const v2f*)(xp + 12);
        if (c + 4 < NCHUNK) __builtin_prefetch(xp + 64, 0, 1);   // global_prefetch_b8

        v8f acc;
        #pragma unroll
        for (int j = 0; j < 8; ++j) acc[j] = cf[j] * carry_bc;   // carry term (0 for c==0)
        acc = wmma16x16x16((c == 0) ? A0m : Afwd, B0, B1, B2, B3, acc);

        // new carry = y[15] : D vgpr7, lanes 16..31, N = lane-16 -> broadcast per N
        carry_bc = __shfl(acc[7], 16 + n, 32);
        if (hi == 0) carries[wave][c + 1][n] = carry_bc;         // carry entering chunk c+1
    }

    // ============ backward pass: recompute y, reverse EMA, PCEN ===========
    float* zt = ztile[wave];
    float bcarry = 0.0f;
    for (int c = NCHUNK - 1; c >= 0; --c) {
        const int t0 = c * 16;
        const float* xp = x + xrow + t0 + 2 * hi;
        v2f B0 = *(const v2f*)(xp + 0);
        v2f B1 = *(const v2f*)(xp + 4);
        v2f B2 = *(const v2f*)(xp + 8);
        v2f B3 = *(const v2f*)(xp + 12);

        // recompute forward chunk y (x is L2-resident: 16 MB << 192 MB L2)
        const float cin = carries[wave][c][n];
        v8f y;
        #pragma unroll
        for (int j = 0; j < 8; ++j) y[j] = (c == 0) ? 0.0f : cf[j] * cin;
        y = wmma16x16x16((c == 0) ? A0m : Afwd, B0, B1, B2, B3, y);

        // route y from D-layout to B-layout via LDS (row stride 20 dwords)
        #pragma unroll
        for (int j = 0; j < 8; ++j) zt[n * 20 + j + 8 * hi] = y[j];
        asm volatile("s_wait_dscnt 0" ::: "memory");  // cross-lane LDS RAW inside wave
        const float* zr = zt + n * 20 + 2 * hi;
        v2f Y0 = *(const v2f*)(zr + 0);
        v2f Y1 = *(const v2f*)(zr + 4);
        v2f Y2 = *(const v2f*)(zr + 8);
        v2f Y3 = *(const v2f*)(zr + 12);

        v8f z;
        #pragma unroll
        for (int j = 0; j < 8; ++j) z[j] = (c == NCHUNK - 1) ? 0.0f : cr[j] * bcarry;
        z = wmma16x16x16((c == NCHUNK - 1) ? ArevL : Arev, Y0, Y1, Y2, Y3, z);

        // next backward carry = z[0] (c' at t0): D vgpr0, lanes 0..15
        bcarry = __shfl(z[0], n, 32);

        // fused PCEN tail + store
        #pragma unroll
        for (int j = 0; j < 8; ++j) {
            const int t = t0 + j + 8 * hi;
            const float xv = x[xrow + t];
            const float Mp = __expf(-alpha * (lge + log1pf(z[j] * inve)));  // (eps+M)^-alpha
            out[orow + t] = __powf(fmaf(xv, Mp, delta), r) - dr;
        }
    }
}

extern "C" void kernel_launch(void* const* d_in, const int* in_sizes, int n_in,
                              void* d_out, int out_size, void* d_ws, size_t ws_size,
                              hipStream_t stream) {
    const float* x       = (const float*)d_in[0];
    const float* i_alpha = (const float*)d_in[1];
    const float* logdel  = (const float*)d_in[2];
    const float* i_r     = (const float*)d_in[3];
    const float* i_s     = (const float*)d_in[4];
    const float* i_eps   = (const float*)d_in[5];
    float* out           = (float*)d_out;
    (void)in_sizes; (void)n_in; (void)out_size; (void)d_ws; (void)ws_size;

    dim3 grid(NTILES / WPB);   // 256 blocks
    dim3 block(WPB * 32);      // 128 threads = 4 waves
    pcen_wmma_kernel<<<grid, block, 0, stream>>>(x, i_alpha, logdel, i_r, i_s, i_eps, out);
}